// DGCNN_42958262895007
// MI455X (gfx1250) — compile-verified
//
#include <hip/hip_runtime.h>

#define BB   8
#define NN   4096
#define KNB  16

typedef __attribute__((ext_vector_type(2))) float v2f;
typedef __attribute__((ext_vector_type(8))) float v8f;

__device__ __forceinline__ v8f wmma_f32_16x16x4(v2f a, v2f b, v8f c) {
  // D = A(16x4) * B(4x16) + C, exact fp32
  return __builtin_amdgcn_wmma_f32_16x16x4_f32(false, a, false, b, (short)0, c, false, false);
}

// ---------------------------------------------------------------- utilities
__global__ void zero_kernel(float* __restrict__ p, int n) {
  int i = blockIdx.x * blockDim.x + threadIdx.x;
  if (i < n) p[i] = 0.0f;
}

// xt1[b,n,c] (stride 4, c=3 zero-padded) and sq1[b,n] from x (B,3,N)
__global__ void make_xt1(const float* __restrict__ x, float* __restrict__ xt,
                         float* __restrict__ sq) {
  int p = blockIdx.x * blockDim.x + threadIdx.x;   // p = b*N + n
  if (p < BB * NN) {
    int b = p / NN, n = p % NN;
    float s = 0.f;
    #pragma unroll
    for (int c = 0; c < 3; ++c) {
      float v = x[((size_t)b * 3 + c) * NN + n];
      xt[(size_t)p * 4 + c] = v;
      s += v * v;
    }
    xt[(size_t)p * 4 + 3] = 0.f;
    sq[p] = s;
  }
}

// Acomb = [Wl - Wr | Wr], each half padded to CH columns. W is (COUT, 2*CINh).
__global__ void prep_acomb(const float* __restrict__ W, float* __restrict__ A,
                           int CINh, int CH, int COUT) {
  int i = blockIdx.x * blockDim.x + threadIdx.x;
  if (i < COUT * 2 * CH) {
    int o = i / (2 * CH), c = i % (2 * CH);
    float v;
    if (c < CH) {
      v = (c < CINh) ? (W[o * 2 * CINh + c] - W[o * 2 * CINh + CINh + c]) : 0.f;
    } else {
      int cc = c - CH;
      v = (cc < CINh) ? W[o * 2 * CINh + CINh + cc] : 0.f;
    }
    A[i] = v;
  }
}

__global__ void bn_finalize(const float* __restrict__ sums,
                            const float* __restrict__ gamma,
                            const float* __restrict__ beta,
                            float* __restrict__ bn_ab, int cout, float inv_count) {
  int o = blockIdx.x * blockDim.x + threadIdx.x;
  if (o < cout) {
    float mean = sums[o] * inv_count;
    float var  = sums[cout + o] * inv_count - mean * mean;
    float a = gamma[o] * rsqrtf(var + 1e-5f);
    bn_ab[o] = a;
    bn_ab[cout + o] = beta[o] - a * mean;
  }
}

// ---------------------------------------------------------------- kNN (top-16)
// One wave owns 32 rows (two WMMA M-strips): B-tile is loaded once and used by
// both strips; all 32 lanes then own one row each for the register top-16 scan.
template <int C>  // channel stride: 4 (layer1, padded) or 32
__global__ void __launch_bounds__(32)
knn_kernel(const float* __restrict__ xt, const float* __restrict__ sq,
           int* __restrict__ idx) {
  const int b    = blockIdx.y;
  const int m0   = blockIdx.x * 32;
  const int lane = threadIdx.x;
  const int lm   = lane & 15;
  const int half = lane >> 4;
  const float* xb  = xt + (size_t)b * NN * C;
  const float* sqb = sq + (size_t)b * NN;

  __shared__ float pdl[32][17];

  // A matrices for the two strips (persistent): lane holds M=lm, K=kc+2*half+v
  v2f a0[C / 4], a1[C / 4];
  #pragma unroll
  for (int kc4 = 0; kc4 < C / 4; ++kc4) {
    const float* p0 = xb + (size_t)(m0 + lm) * C + kc4 * 4 + half * 2;
    a0[kc4].x = p0[0]; a0[kc4].y = p0[1];
    const float* p1 = xb + (size_t)(m0 + 16 + lm) * C + kc4 * 4 + half * 2;
    a1[kc4].x = p1[0]; a1[kc4].y = p1[1];
  }
  // |x_row|^2 for the rows this lane produces in the D tiles (loop-invariant)
  float srow0[8], srow1[8];
  #pragma unroll
  for (int r = 0; r < 8; ++r) {
    srow0[r] = sqb[m0 + r + half * 8];
    srow1[r] = sqb[m0 + 16 + r + half * 8];
  }

  float vals[16]; int inds[16];
  #pragma unroll
  for (int i = 0; i < 16; ++i) { vals[i] = -3.4e38f; inds[i] = 0; }
  float vmin = -3.4e38f;

  for (int t = 0; t < NN / 16; ++t) {
    const int j0 = t * 16;
    v8f acc0 = {}, acc1 = {};
    #pragma unroll
    for (int kc4 = 0; kc4 < C / 4; ++kc4) {
      const float* p = xb + (size_t)(j0 + lm) * C + kc4 * 4 + half * 2;
      v2f bv; bv.x = p[0]; bv.y = p[1];
      acc0 = wmma_f32_16x16x4(a0[kc4], bv, acc0);
      acc1 = wmma_f32_16x16x4(a1[kc4], bv, acc1);
    }
    const float scol = sqb[j0 + lm];
    #pragma unroll
    for (int r = 0; r < 8; ++r) {        // D layout: M = r + 8*half, N = lm
      pdl[r + half * 8][lm]      = 2.0f * acc0[r] - srow0[r] - scol;
      pdl[16 + r + half * 8][lm] = 2.0f * acc1[r] - srow1[r] - scol;
    }
    __syncthreads();
    // every lane owns row (m0 + lane)
    #pragma unroll
    for (int c = 0; c < 16; ++c) {
      const float v = pdl[lane][c];
      if (v > vmin) {
        int slot = 0; float mv = vals[0];
        #pragma unroll
        for (int i = 1; i < 16; ++i) if (vals[i] < mv) { mv = vals[i]; slot = i; }
        #pragma unroll
        for (int i = 0; i < 16; ++i) if (i == slot) { vals[i] = v; inds[i] = j0 + c; }
        vmin = vals[0];
        #pragma unroll
        for (int i = 1; i < 16; ++i) vmin = fminf(vmin, vals[i]);
      }
    }
    __syncthreads();
  }

  int* op = idx + ((size_t)b * NN + m0 + lane) * KNB;
  #pragma unroll
  for (int i = 0; i < 16; ++i) op[i] = inds[i];
}

// ---------------------------------------------------------------- EdgeConv
// One wave handles one point's 16 edges (= one WMMA M-tile).
// mode 0: accumulate per-channel sum / sumsq (BN batch stats).
// mode 1: y = a*acc + b, ReLU, max over k, store features / sq / global max.
template <int CH, int COUT>  // CH = padded per-half channels (4|32)
__global__ void __launch_bounds__(256)
edge_conv(const float* __restrict__ xt, const int* __restrict__ idx,
          const float* __restrict__ Ac, int mode, int pts_per_wave,
          const float* __restrict__ bn_ab, float* __restrict__ sums,
          float* __restrict__ x_out, float* __restrict__ sq_out,
          unsigned* __restrict__ gmax, int gbase) {
  const int lane = threadIdx.x & 31;
  const int wave = (blockIdx.x * blockDim.x + threadIdx.x) >> 5;
  const int lm   = lane & 15;
  const int half = lane >> 4;
  constexpr int KT = 2 * CH;
  constexpr int NT = COUT / 16;

  // Weight (B-matrix) tiles are point-invariant: hoist into registers.
  v2f breg[KT / 4][NT];
  #pragma unroll
  for (int kc4 = 0; kc4 < KT / 4; ++kc4)
    #pragma unroll
    for (int t = 0; t < NT; ++t) {
      const float* bp = Ac + (size_t)(t * 16 + lm) * KT + kc4 * 4 + half * 2;
      breg[kc4][t].x = bp[0];
      breg[kc4][t].y = bp[1];
    }

  float ssum[NT], ssq[NT];
  #pragma unroll
  for (int t = 0; t < NT; ++t) { ssum[t] = 0.f; ssq[t] = 0.f; }

  const v8f vzero = {};

  for (int pi = 0; pi < pts_per_wave; ++pi) {
    const int p = wave * pts_per_wave + pi;     // p = b*N + n
    const int b = p / NN;
    const int nb = idx[(size_t)p * KNB + lm];   // lane owns edge row M = lm
    const float* xc = xt + (size_t)p * CH;
    const float* xn = xt + ((size_t)b * NN + nb) * CH;

    v8f acc[NT];
    #pragma unroll
    for (int t = 0; t < NT; ++t) acc[t] = vzero;

    #pragma unroll
    for (int kc4 = 0; kc4 < KT / 4; ++kc4) {
      const int kc = kc4 * 4;
      // A row m = [ x_n (CH) | x_nb (CH) ], chunk is statically center/neighbor
      const float* src = (kc < CH) ? (xc + kc) : (xn + (kc - CH));
      v2f av; av.x = src[half * 2]; av.y = src[half * 2 + 1];
      #pragma unroll
      for (int t = 0; t < NT; ++t)
        acc[t] = wmma_f32_16x16x4(av, breg[kc4][t], acc[t]);
    }

    if (mode == 0) {
      #pragma unroll
      for (int t = 0; t < NT; ++t)
        #pragma unroll
        for (int r = 0; r < 8; ++r) {
          const float y = acc[t][r];
          ssum[t] += y; ssq[t] += y * y;
        }
    } else {
      float sqp = 0.f;
      #pragma unroll
      for (int t = 0; t < NT; ++t) {
        const int ch = t * 16 + lm;
        const float a  = bn_ab[ch];
        const float bc = bn_ab[COUT + ch];
        float m = 0.f;
        #pragma unroll
        for (int r = 0; r < 8; ++r) {
          float y = fmaf(a, acc[t][r], bc);
          y = fmaxf(y, 0.f);              // relu∘leaky_relu == relu
          m = fmaxf(m, y);                // max over edge rows in this half
        }
        m = fmaxf(m, __shfl_xor(m, 16, 32));  // combine lane halves (full k-max)
        if (lane < 16) {
          if (x_out) x_out[(size_t)p * COUT + ch] = m;
          atomicMax(&gmax[(size_t)b * 128 + gbase + ch], __float_as_uint(m));
          sqp += m * m;
        }
      }
      if (sq_out) {
        sqp += __shfl_xor(sqp, 8, 32);
        sqp += __shfl_xor(sqp, 4, 32);
        sqp += __shfl_xor(sqp, 2, 32);
        sqp += __shfl_xor(sqp, 1, 32);
        if (lane == 0) sq_out[p] = sqp;
      }
    }
  }

  if (mode == 0) {
    #pragma unroll
    for (int t = 0; t < NT; ++t) {
      ssum[t] += __shfl_xor(ssum[t], 16, 32);
      ssq[t]  += __shfl_xor(ssq[t], 16, 32);
      if (lane < 16) {
        atomicAdd(&sums[t * 16 + lm], ssum[t]);
        atomicAdd(&sums[COUT + t * 16 + lm], ssq[t]);
      }
    }
  }
}

// ---------------------------------------------------------------- final FC
__global__ void fc_kernel(const unsigned* __restrict__ gmax,
                          const float* __restrict__ Wfc,
                          const float* __restrict__ bfc,
                          float* __restrict__ out) {
  const int i = threadIdx.x;       // 512 threads: b*64 + o
  const int b = i >> 6, o = i & 63;
  float s = bfc[o];
  #pragma unroll 4
  for (int c = 0; c < 128; ++c)
    s += __uint_as_float(gmax[b * 128 + c]) * Wfc[o * 128 + c];
  out[i] = s;
}

// ---------------------------------------------------------------- launch
extern "C" void kernel_launch(void* const* d_in, const int* in_sizes, int n_in,
                              void* d_out, int out_size, void* d_ws, size_t ws_size,
                              hipStream_t stream) {
  const float* x   = (const float*)d_in[0];
  const float* W1  = (const float*)d_in[1];
  const float* g1  = (const float*)d_in[2];
  const float* b1  = (const float*)d_in[3];
  const float* W2  = (const float*)d_in[4];
  const float* g2  = (const float*)d_in[5];
  const float* b2  = (const float*)d_in[6];
  const float* W3  = (const float*)d_in[7];
  const float* g3  = (const float*)d_in[8];
  const float* b3  = (const float*)d_in[9];
  const float* Wfc = (const float*)d_in[10];
  const float* bfc = (const float*)d_in[11];
  float* out = (float*)d_out;

  char* ws = (char*)d_ws;
  size_t off = 0;
  auto alloc = [&](size_t bytes) -> void* {
    void* p = ws + off;
    off = (off + bytes + 255) & ~(size_t)255;
    return p;
  };

  float* xt1  = (float*)alloc((size_t)BB * NN * 4 * 4);
  float* sq1  = (float*)alloc((size_t)BB * NN * 4);
  int*   idx1 = (int*)  alloc((size_t)BB * NN * KNB * 4);
  float* xt2  = (float*)alloc((size_t)BB * NN * 32 * 4);
  float* sq2  = (float*)alloc((size_t)BB * NN * 4);
  int*   idx2 = (int*)  alloc((size_t)BB * NN * KNB * 4);
  float* xt3  = (float*)alloc((size_t)BB * NN * 32 * 4);
  float* sq3  = (float*)alloc((size_t)BB * NN * 4);
  int*   idx3 = (int*)  alloc((size_t)BB * NN * KNB * 4);
  float* ac1  = (float*)alloc(32 * 8 * 4);
  float* ac2  = (float*)alloc(32 * 64 * 4);
  float* ac3  = (float*)alloc(64 * 64 * 4);
  float* sums1 = (float*)alloc(2 * 32 * 4);
  float* sums2 = (float*)alloc(2 * 32 * 4);
  float* sums3 = (float*)alloc(2 * 64 * 4);
  float* ab1  = (float*)alloc(2 * 32 * 4);
  float* ab2  = (float*)alloc(2 * 32 * 4);
  float* ab3  = (float*)alloc(2 * 64 * 4);
  unsigned* gmax = (unsigned*)alloc((size_t)BB * 128 * 4);

  const float invc = 1.0f / ((float)BB * NN * KNB);

  // init + prep
  zero_kernel<<<(BB * 128 + 255) / 256, 256, 0, stream>>>((float*)gmax, BB * 128);
  zero_kernel<<<1, 256, 0, stream>>>(sums1, 64);
  zero_kernel<<<1, 256, 0, stream>>>(sums2, 64);
  zero_kernel<<<1, 256, 0, stream>>>(sums3, 128);
  make_xt1<<<(BB * NN + 255) / 256, 256, 0, stream>>>(x, xt1, sq1);
  prep_acomb<<<1, 256, 0, stream>>>(W1, ac1, 3, 4, 32);
  prep_acomb<<<(32 * 64 + 255) / 256, 256, 0, stream>>>(W2, ac2, 32, 32, 32);
  prep_acomb<<<(64 * 64 + 255) / 256, 256, 0, stream>>>(W3, ac3, 32, 32, 64);

  // ---- layer 1: C=3(pad 4) -> 32
  knn_kernel<4><<<dim3(NN / 32, BB), 32, 0, stream>>>(xt1, sq1, idx1);
  edge_conv<4, 32><<<512, 256, 0, stream>>>(xt1, idx1, ac1, 0, 8, nullptr, sums1,
                                            nullptr, nullptr, gmax, 0);
  bn_finalize<<<1, 32, 0, stream>>>(sums1, g1, b1, ab1, 32, invc);
  edge_conv<4, 32><<<4096, 256, 0, stream>>>(xt1, idx1, ac1, 1, 1, ab1, nullptr,
                                             xt2, sq2, gmax, 0);

  // ---- layer 2: 32 -> 32
  knn_kernel<32><<<dim3(NN / 32, BB), 32, 0, stream>>>(xt2, sq2, idx2);
  edge_conv<32, 32><<<512, 256, 0, stream>>>(xt2, idx2, ac2, 0, 8, nullptr, sums2,
                                             nullptr, nullptr, gmax, 32);
  bn_finalize<<<1, 32, 0, stream>>>(sums2, g2, b2, ab2, 32, invc);
  edge_conv<32, 32><<<4096, 256, 0, stream>>>(xt2, idx2, ac2, 1, 1, ab2, nullptr,
                                              xt3, sq3, gmax, 32);

  // ---- layer 3: 32 -> 64
  knn_kernel<32><<<dim3(NN / 32, BB), 32, 0, stream>>>(xt3, sq3, idx3);
  edge_conv<32, 64><<<512, 256, 0, stream>>>(xt3, idx3, ac3, 0, 8, nullptr, sums3,
                                             nullptr, nullptr, gmax, 64);
  bn_finalize<<<1, 64, 0, stream>>>(sums3, g3, b3, ab3, 64, invc);
  edge_conv<32, 64><<<4096, 256, 0, stream>>>(xt3, idx3, ac3, 1, 1, ab3, nullptr,
                                              nullptr, nullptr, gmax, 64);

  // ---- FC 128 -> 64
  fc_kernel<<<1, 512, 0, stream>>>(gmax, Wfc, bfc, out);
}